// PriorMix_57939108823098
// MI455X (gfx1250) — compile-verified
//
#include <hip/hip_runtime.h>
#include <hip/hip_bf16.h>
#include <math.h>

// Problem constants (from reference)
#define B_SZ        512
#define NRBOX       18
#define D_DIM       2048
#define NR_OBJ      16
#define KK          3
#define NUM_CLASSES 174
#define NUM_CAT     321
#define NCAND       (B_SZ * NR_OBJ)     // 8192 candidates per batch row
#define CAT_WORDS   (B_SZ * NRBOX)      // 9216 ints = 36 KB

typedef __attribute__((ext_vector_type(2))) float v2f;
typedef __attribute__((ext_vector_type(8))) float v8f;

struct PmParams {
    int   sel_b[KK];   // selected source batch
    int   sel_o[KK];   // selected source box (0..15)
    int   slot[KK];    // own slot (0..15)
    int   has_cand;
    float lam;
    int   pad;
};                     // 48 bytes

// ---------------- deterministic counter-based RNG ----------------
__device__ __forceinline__ unsigned pm_mix32(unsigned x) {
    x ^= x >> 16; x *= 0x7feb352du;
    x ^= x >> 15; x *= 0x846ca68bu;
    x ^= x >> 16;
    return x;
}
__device__ __forceinline__ unsigned pm_rng3(unsigned s, unsigned a, unsigned b) {
    return pm_mix32(s ^ pm_mix32(a ^ pm_mix32(b + 0x9E3779B9u)));
}

// ---------------------------------------------------------------------------
// Kernel A: per-batch row reductions via f32 WMMA.
//   ori_comp[i]  = mean over all 18 boxes
//   objs_mean[i] = mean over boxes 2..17 (fallback features)
// All-ones 16x4 A fragment -> D[m,n] = sum_k B[k,n] for every m, so each
// lane's acc[0] is the column sum of its 16-wide tile. Rows 0..15 go through
// a 4-deep WMMA chain (branch-free); rows 16/17 are folded in with scalar
// adds, eliminating the exec-mask guarded tail loads of the previous version.
// ---------------------------------------------------------------------------
__global__ __launch_bounds__(256) void pm_mean(const float* __restrict__ obj_fea,
                                               float* __restrict__ ori_comp,
                                               float* __restrict__ objs_mean)
{
    const int i    = blockIdx.x;
    const int wave = threadIdx.x >> 5;     // 8 waves
    const int lane = threadIdx.x & 31;
    const int col  = lane & 15;
    const int rsel = (lane >> 4) << 1;     // lanes 0-15 -> rows +0,+1 ; lanes 16-31 -> +2,+3
    const float* base = obj_fea + (size_t)i * NRBOX * D_DIM;

    v2f a; a[0] = 1.0f; a[1] = 1.0f;       // all-ones 16x4 A fragment

    for (int t = wave; t < D_DIM / 16; t += 8) {
        const int c0 = t * 16;
        v8f acc = {};
#pragma unroll
        for (int g = 0; g < 4; ++g) {      // rows 0..15, branch-free
            const int r0 = g * 4 + rsel;
            v2f b;
            b[0] = base[(size_t)(r0 + 0) * D_DIM + c0 + col];
            b[1] = base[(size_t)(r0 + 1) * D_DIM + c0 + col];
            acc = __builtin_amdgcn_wmma_f32_16x16x4_f32(
                false, a, false, b, (short)0, acc, false, false);
        }
        const float r16  = base[(size_t)16 * D_DIM + c0 + col];
        const float r17  = base[(size_t)17 * D_DIM + c0 + col];
        const float full = acc[0] + r16 + r17;                      // sum rows 0..17
        const float r01  = base[c0 + col] + base[D_DIM + c0 + col]; // rows 0+1 (cache hot)
        if (lane < 16) {
            ori_comp [(size_t)i * D_DIM + c0 + col] = full * (1.0f / 18.0f);
            objs_mean[(size_t)i * D_DIM + c0 + col] = (full - r01) * (1.0f / 16.0f);
        }
    }
}

// ---------------------------------------------------------------------------
// Kernel B: fused co-occurrence gather + log-weights + Gumbel argmax (x3),
// has_cand, lambda, own-slot picks.
// The 36 KB obj_category table is staged into LDS once per block with
// gfx1250 async global->LDS b128 loads (ASYNCcnt-tracked, no VGPR landing
// zone), so the 8192-candidate loop gathers categories at LDS latency
// instead of making 4.2M L2 round-trips across the grid.
// ---------------------------------------------------------------------------
__global__ __launch_bounds__(256) void pm_sample(const float* __restrict__ cooc,
                                                 const int*   __restrict__ obj_indicator,
                                                 const int*   __restrict__ obj_category,
                                                 const int*   __restrict__ labels,
                                                 PmParams*    __restrict__ params)
{
    const int i   = blockIdx.x;
    const int tid = threadIdx.x;

    __shared__ int   s_cat[CAT_WORDS];     // 36 KB
    __shared__ float sv[KK][256];
    __shared__ int   si[KK][256];
    __shared__ int   s_any;
    if (tid == 0) s_any = 0;

    // --- async stage: global obj_category -> LDS, b128 per issue -----------
    {
        const unsigned lds_base = (unsigned)(size_t)&s_cat[0];
        for (int t = tid; t < CAT_WORDS / 4; t += 256) {
            const unsigned lds_addr = lds_base + (unsigned)t * 16u;
            const int*     gaddr    = obj_category + (size_t)t * 4;
            asm volatile("global_load_async_to_lds_b128 %0, %1, off"
                         :: "v"(lds_addr), "v"(gaddr) : "memory");
        }
        asm volatile("s_wait_asynccnt 0x0" ::: "memory");
    }
    __syncthreads();

    const int lab = labels[i];
    const float* wrow = cooc + (size_t)lab * NUM_CAT;

    float bestv[KK];
    int   besti[KK];
#pragma unroll
    for (int k = 0; k < KK; ++k) { bestv[k] = -INFINITY; besti[k] = 0; }
    int anypos = 0;

    for (int j = tid; j < NCAND; j += 256) {
        const int jb = j >> 4;
        const int jo = j & 15;
        float w = 0.0f;
        if (jb != i) {
            const int cat = s_cat[jb * NRBOX + 2 + jo];
            w = wrow[cat];
        }
        float logit;
        if (w > 0.0f) { logit = __logf(w); anypos = 1; }
        else          { logit = -INFINITY; }

#pragma unroll
        for (int k = 0; k < KK; ++k) {
            const unsigned h = pm_rng3(0x1337C0DEu + (unsigned)k, (unsigned)i, (unsigned)j);
            const float u = (float)(h >> 9) * 1.1920929e-7f + 5.9604645e-8f;  // (0,1)
            const float g = -__logf(-__logf(u));                              // Gumbel
            const float v = logit + g;
            if (v > bestv[k]) { bestv[k] = v; besti[k] = j; }
        }
    }

    if (anypos) atomicOr(&s_any, 1);
    if (tid < NR_OBJ) {
        if (obj_indicator[i * NRBOX + 2 + tid] != 0) atomicOr(&s_any, 2);
    }
#pragma unroll
    for (int k = 0; k < KK; ++k) { sv[k][tid] = bestv[k]; si[k][tid] = besti[k]; }
    __syncthreads();

    for (int off = 128; off > 0; off >>= 1) {
        if (tid < off) {
#pragma unroll
            for (int k = 0; k < KK; ++k) {
                const float vo = sv[k][tid + off];
                const int   io = si[k][tid + off];
                if (vo > sv[k][tid] || (vo == sv[k][tid] && io < si[k][tid])) {
                    sv[k][tid] = vo; si[k][tid] = io;
                }
            }
        }
        __syncthreads();
    }

    if (tid == 0) {
        PmParams p;
#pragma unroll
        for (int k = 0; k < KK; ++k) {
            const int s = si[k][0];
            p.sel_b[k] = s >> 4;
            p.sel_o[k] = s & 15;
            p.slot[k]  = (int)(pm_rng3(0xABCD1234u, (unsigned)i, (unsigned)k) & 15u);
        }
        p.has_cand = (s_any == 3) ? 1 : 0;
        const unsigned hl = pm_rng3(0x5EED5EEDu, (unsigned)i, 0u);
        p.lam = (float)(hl >> 8) * 5.9604645e-8f;   // [0,1)
        p.pad = 0;
        params[i] = p;
    }
}

// ---------------------------------------------------------------------------
// Kernel C: gather + mix. One block per (i,k); b128 loads; lambda-blend or
// objs_mean fallback; one-hot label blend.
// ---------------------------------------------------------------------------
__global__ __launch_bounds__(256) void pm_mixk(const float*    __restrict__ obj_fea,
                                               const int*      __restrict__ labels,
                                               const float*    __restrict__ objs_mean,
                                               const PmParams* __restrict__ params,
                                               float* __restrict__ mix_fea,
                                               float* __restrict__ mix_label)
{
    const int i = blockIdx.x;
    const int k = blockIdx.y;
    const PmParams p = params[i];
    const float lam = p.lam;
    const float oml = 1.0f - lam;
    const int   hc  = p.has_cand;

    const float* own = obj_fea + ((size_t)i * NRBOX + 2 + p.slot[k]) * D_DIM;
    const float* sel = obj_fea + ((size_t)p.sel_b[k] * NRBOX + 2 + p.sel_o[k]) * D_DIM;
    const float* fb  = objs_mean + (size_t)i * D_DIM;
    float* out = mix_fea + ((size_t)i * KK + k) * D_DIM;

    __builtin_prefetch(sel, 0, 1);     // global_prefetch_b8
    __builtin_prefetch(own, 0, 1);

    const float4* own4 = (const float4*)own;
    const float4* sel4 = (const float4*)sel;
    const float4* fb4  = (const float4*)fb;
    float4*       out4 = (float4*)out;

    for (int c = threadIdx.x; c < D_DIM / 4; c += 256) {
        const float4 a = own4[c];
        const float4 b = sel4[c];
        const float4 f = fb4[c];
        float4 r;
        r.x = hc ? a.x * lam + b.x * oml : f.x;
        r.y = hc ? a.y * lam + b.y * oml : f.y;
        r.z = hc ? a.z * lam + b.z * oml : f.z;
        r.w = hc ? a.w * lam + b.w * oml : f.w;
        out4[c] = r;
    }

    const int li = labels[i];
    const int lj = labels[p.sel_b[k]];
    float* lout = mix_label + ((size_t)i * KK + k) * NUM_CLASSES;
    for (int c = threadIdx.x; c < NUM_CLASSES; c += 256) {
        const float a = (c == li) ? 1.0f : 0.0f;
        const float b = (c == lj) ? 1.0f : 0.0f;
        lout[c] = hc ? a * lam + b * oml : a;
    }
}

// ---------------------------------------------------------------------------
extern "C" void kernel_launch(void* const* d_in, const int* in_sizes, int n_in,
                              void* d_out, int out_size, void* d_ws, size_t ws_size,
                              hipStream_t stream)
{
    (void)in_sizes; (void)n_in; (void)out_size; (void)ws_size;

    const float* obj_fea       = (const float*)d_in[0];
    const float* cooccurrence  = (const float*)d_in[1];
    const int*   obj_indicator = (const int*)  d_in[2];
    const int*   obj_category  = (const int*)  d_in[3];
    const int*   labels        = (const int*)  d_in[4];

    float* out       = (float*)d_out;
    float* ori_comp  = out;                                         // 512*2048
    float* mix_fea   = out + (size_t)B_SZ * D_DIM;                  // 512*3*2048
    float* mix_label = mix_fea + (size_t)B_SZ * KK * D_DIM;         // 512*3*174

    float*    objs_mean = (float*)d_ws;                             // 4 MB
    PmParams* params    = (PmParams*)((char*)d_ws + (size_t)B_SZ * D_DIM * sizeof(float));

    pm_mean  <<<B_SZ, 256, 0, stream>>>(obj_fea, ori_comp, objs_mean);
    pm_sample<<<B_SZ, 256, 0, stream>>>(cooccurrence, obj_indicator, obj_category,
                                        labels, params);
    pm_mixk  <<<dim3(B_SZ, KK), 256, 0, stream>>>(obj_fea, labels, objs_mean,
                                                  params, mix_fea, mix_label);
}